// PointNet_68874095559314
// MI455X (gfx1250) — compile-verified
//
#include <hip/hip_runtime.h>
#include <hip/hip_bf16.h>

// ---------------- problem constants (match reference) ----------------
#define BB    8
#define NPTS  16384
#define SS    4096
#define CFEAT 9

typedef float v2f __attribute__((ext_vector_type(2)));
typedef float v8f __attribute__((ext_vector_type(8)));

// TDM descriptor group types (amdgpu-toolchain clang-23: 6-arg builtin form)
typedef unsigned int tdm_v4u __attribute__((ext_vector_type(4)));
typedef int          tdm_v8i __attribute__((ext_vector_type(8)));
typedef int          tdm_v4i __attribute__((ext_vector_type(4)));

// ----------------------------------------------------------------------
// Wave-level GEMM microkernel:  out(16xN) = relu(in(16xK) @ W(KxN) + bias)
// Uses V_WMMA_F32_16X16X4_F32.  K multiple of 4, N multiple of 16.
// Fragment layouts per CDNA5 ISA 7.12.2:
//   A 16x4 : lane l -> row M = l%16 ; VGPR{0,1} hold K = 2*(l/16)+{0,1}
//   B 4x16 : lane l -> col N = l%16 ; VGPR{0,1} hold K = 2*(l/16)+{0,1}
//   C/D    : VGPR r, lane l -> (M = r + 8*(l/16), N = l%16)
// ----------------------------------------------------------------------
static __device__ __forceinline__ void wave_gemm_relu(
    const float* __restrict__ W, const float* __restrict__ bias,
    const float* __restrict__ inLDS, int sin,
    float* __restrict__ outLDS, int sout,
    int Kdim, int Ndim, int lane)
{
    const int half = lane >> 4;      // 0: lanes 0-15, 1: lanes 16-31
    const int l16  = lane & 15;
    for (int n0 = 0; n0 < Ndim; n0 += 16) {
        v8f c = {};
        for (int k0 = 0; k0 < Kdim; k0 += 4) {
            const int ka = k0 + 2 * half;
            v2f a, b;
            a.x = inLDS[l16 * sin + ka];
            a.y = inLDS[l16 * sin + ka + 1];
            b.x = W[(ka)     * Ndim + n0 + l16];
            b.y = W[(ka + 1) * Ndim + n0 + l16];
            c = __builtin_amdgcn_wmma_f32_16x16x4_f32(
                    false, a, false, b, (short)0, c, false, false);
        }
        const float bb = bias[n0 + l16];
#pragma unroll
        for (int r = 0; r < 8; ++r) {
            float v = c[r] + bb;
            outLDS[(r + 8 * half) * sout + n0 + l16] = v > 0.f ? v : 0.f;
        }
    }
}

// ---------------- kernel 0: zero the context-max accumulator ----------
__global__ void init_ctx_kernel(unsigned* ctx, int n)
{
    int i = blockIdx.x * blockDim.x + threadIdx.x;
    if (i < n) ctx[i] = 0u;   // 0u == 0.0f; relu outputs are >= 0
}

// ---------------- kernel 1: mlp1 + mlp2 + context max ----------------
// 2 waves / block, each wave owns a 16-point tile.
#define ENC_SA 260   // 260 % 64 == 4 -> conflict-free column reads
#define ENC_SB 132
__global__ __launch_bounds__(64) void encoder_kernel(
    const float* __restrict__ x, const float* __restrict__ pos,
    const float* __restrict__ W10, const float* __restrict__ b10,
    const float* __restrict__ W11, const float* __restrict__ b11,
    const float* __restrict__ W20, const float* __restrict__ b20,
    const float* __restrict__ W21, const float* __restrict__ b21,
    float* __restrict__ f1_ws, unsigned* __restrict__ ctx)
{
    __shared__ float bufA[2][16 * ENC_SA];   // widest activation: 256
    __shared__ float bufB[2][16 * ENC_SB];   // widest activation: 128

    const int wave = threadIdx.x >> 5;
    const int lane = threadIdx.x & 31;
    const int tile = blockIdx.x * 2 + wave;          // 16-point tile id
    const int b    = tile / (SS / 16);
    const int row0 = (tile % (SS / 16)) * 16;

    // warm L2/WGP$ for the big streamed weights (global_prefetch_b8)
    __builtin_prefetch(W21, 0, 3);
    __builtin_prefetch(W20, 0, 3);

    float* A  = bufA[wave];
    float* Bf = bufB[wave];

    // stage input features [pos_s(3) | x_s(9)] : 16 x 12 (192 = 6*32, uniform)
    for (int idx = lane; idx < 16 * 12; idx += 32) {
        const int r = idx / 12, cc = idx % 12;
        const int p = b * NPTS + row0 + r;
        A[r * ENC_SA + cc] = (cc < 3) ? pos[p * 3 + cc] : x[p * CFEAT + (cc - 3)];
    }
    __builtin_amdgcn_wave_barrier();

    wave_gemm_relu(W10, b10, A,  ENC_SA, Bf, ENC_SB, 12,  64,  lane); // 12 ->64
    __builtin_amdgcn_wave_barrier();
    wave_gemm_relu(W11, b11, Bf, ENC_SB, A,  ENC_SA, 64,  64,  lane); // 64 ->64  (= f1)
    __builtin_amdgcn_wave_barrier();

    // persist f1 tile (needed by mlp3)
    for (int idx = lane; idx < 16 * 64; idx += 32) {
        const int r = idx >> 6, cc = idx & 63;
        f1_ws[(size_t)(b * SS + row0 + r) * 64 + cc] = A[r * ENC_SA + cc];
    }
    __builtin_amdgcn_wave_barrier();

    wave_gemm_relu(W20, b20, A,  ENC_SA, Bf, ENC_SB, 64,  128, lane); // 64 ->128
    __builtin_amdgcn_wave_barrier();
    wave_gemm_relu(W21, b21, Bf, ENC_SB, A,  ENC_SA, 128, 256, lane); // 128->256 (= f2)
    __builtin_amdgcn_wave_barrier();

    // per-batch channel max over this 16-row tile (relu >= 0 -> uint order == float order)
    for (int cc = lane; cc < 256; cc += 32) {
        float m = 0.f;
#pragma unroll
        for (int r = 0; r < 16; ++r) m = fmaxf(m, A[r * ENC_SA + cc]);
        atomicMax(&ctx[b * 256 + cc], __float_as_uint(m));
    }
}

// ---------------- kernel 2: mlp3 + final linear ----------------------
#define HD_SA 324    // holds 320-wide concat input; 324 % 64 == 4
#define HD_SB 132
__global__ __launch_bounds__(64) void head_kernel(
    const float* __restrict__ f1_ws, const unsigned* __restrict__ ctx,
    const float* __restrict__ W30, const float* __restrict__ b30,
    const float* __restrict__ W31, const float* __restrict__ b31,
    const float* __restrict__ Wl,  const float* __restrict__ bl,
    float* __restrict__ logits)
{
    __shared__ float bufA[2][16 * HD_SA];
    __shared__ float bufB[2][16 * HD_SB];

    const int wave = threadIdx.x >> 5;
    const int lane = threadIdx.x & 31;
    const int tile = blockIdx.x * 2 + wave;
    const int b    = tile / (SS / 16);
    const int row0 = (tile % (SS / 16)) * 16;

    __builtin_prefetch(W30, 0, 3);

    float* A  = bufA[wave];
    float* Bf = bufB[wave];

    // stage Gf1 = [ctx(256) | f1(64)] : 16 x 320 (5120 = 160*32, uniform)
    for (int idx = lane; idx < 16 * 320; idx += 32) {
        const int r = idx / 320, cc = idx % 320;
        float v = (cc < 256)
                ? __uint_as_float(ctx[b * 256 + cc])
                : f1_ws[(size_t)(b * SS + row0 + r) * 64 + (cc - 256)];
        A[r * HD_SA + cc] = v;
    }
    __builtin_amdgcn_wave_barrier();

    wave_gemm_relu(W30, b30, A,  HD_SA, Bf, HD_SB, 320, 128, lane);   // 320->128
    __builtin_amdgcn_wave_barrier();
    wave_gemm_relu(W31, b31, Bf, HD_SB, A,  HD_SA, 128, 64,  lane);   // 128->64
    __builtin_amdgcn_wave_barrier();

    // linear 64 -> 2 : lane l handles (row = l%16, out-col = l/16)
    const int r = lane & 15, o = lane >> 4;
    float s = bl[o];
#pragma unroll 8
    for (int k = 0; k < 64; ++k) s += A[r * HD_SA + k] * Wl[k * 2 + o];
    logits[(size_t)(b * SS + row0 + r) * 2 + o] = s;
}

// ---------------- TDM helper: 1-D contiguous dword tile -> LDS --------
// Builds a minimal D# (ISA 8.3/8.4): group0 = {count=1, lds_addr,
// global_addr, type=2}; group1 = {data_size=4B, tensor/tile dim0 = n,
// dim1 = 1, stride0 = n}.  groups 2/3 (and the trailing group) zero:
// tile_dim2/3/4 = 0 -> dimensions disabled.
static __device__ __forceinline__ void tdm_load_1d(
    unsigned lds_byte_off, const void* gptr, unsigned ndwords)
{
    const unsigned long long ga = (unsigned long long)(uintptr_t)gptr;
    tdm_v4u g0;
    g0[0] = 1u;                                            // count=1 (user D#)
    g0[1] = lds_byte_off;                                  // lds_addr
    g0[2] = (unsigned)(ga & 0xffffffffu);                  // global_addr[31:0]
    g0[3] = (unsigned)((ga >> 32) & 0x01ffffffu)           // global_addr[56:32]
          | 0x80000000u;                                   // type=2 ("image")
    tdm_v8i g1;
    g1[0] = 0x00020000;                                    // data_size=2 -> 4 bytes
    g1[1] = (int)((ndwords & 0xffffu) << 16);              // tensor_dim0[15:0]
    g1[2] = (int)((ndwords >> 16) & 0xffffu) | (1 << 16);  // dim0[31:16] | tensor_dim1=1
    g1[3] = (int)((ndwords & 0xffffu) << 16);              // dim1[31:16]=0 | tile_dim0=n
    g1[4] = 1;                                             // tile_dim1=1, tile_dim2=0
    g1[5] = (int)ndwords;                                  // tensor_dim0_stride[31:0]
    g1[6] = 0;                                             // stride0[47:32], stride1[15:0]
    g1[7] = 0;                                             // stride1[47:16]
    tdm_v4i gz4 = {0, 0, 0, 0};
    tdm_v8i gz8 = {0, 0, 0, 0, 0, 0, 0, 0};
    __builtin_amdgcn_tensor_load_to_lds(g0, g1, gz4, gz4, gz8, 0);
}

static __device__ __forceinline__ unsigned lds_off(const void* p)
{
    // LDS aperture: addr[31:0] is the byte offset within the wave's LDS
    return (unsigned)(uintptr_t)p;
}

// ---------------- kernel 3: KNN(K=3) inverse-distance interp ---------
// TDM double-buffered staging: wave 0 DMA-streams chunk c+1 while the
// whole block runs the top-3 scan over chunk c (TENSORcnt-gated).
#define KC      1024           // refs per chunk
#define NCHUNK  (SS / KC)      // 4
__global__ __launch_bounds__(256) void knn_kernel(
    const float* __restrict__ pos, const float* __restrict__ logits,
    float* __restrict__ out)
{
    __shared__ float sPos[2][KC * 3];   // AoS xyz  (12 KB each)
    __shared__ float sLog[2][KC * 2];   // AoS l0l1 ( 8 KB each)

    const int b = blockIdx.x / (NPTS / 256);
    const int q = (blockIdx.x % (NPTS / 256)) * 256 + threadIdx.x;

    const float yx = pos[(size_t)(b * NPTS + q) * 3 + 0];
    const float yy = pos[(size_t)(b * NPTS + q) * 3 + 1];
    const float yz = pos[(size_t)(b * NPTS + q) * 3 + 2];

    float bd0 = 3.4e38f, bd1 = 3.4e38f, bd2 = 3.4e38f;
    float g00 = 0.f, g01 = 0.f, g10 = 0.f, g11 = 0.f, g20 = 0.f, g21 = 0.f;

    // prime the pipeline: chunk 0
    if (threadIdx.x < 32) {
        tdm_load_1d(lds_off(&sPos[0][0]), pos    + (size_t)(b * NPTS) * 3, KC * 3);
        tdm_load_1d(lds_off(&sLog[0][0]), logits + (size_t)(b * SS)   * 2, KC * 2);
    }

    for (int c = 0; c < NCHUNK; ++c) {
        const int cur = c & 1;
        if (threadIdx.x < 32) {
            if (c + 1 < NCHUNK) {
                const int nxt = (c + 1) & 1;   // buffer last read in iter c-1 (barrier-protected)
                tdm_load_1d(lds_off(&sPos[nxt][0]),
                            pos    + ((size_t)b * NPTS + (size_t)(c + 1) * KC) * 3, KC * 3);
                tdm_load_1d(lds_off(&sLog[nxt][0]),
                            logits + ((size_t)b * SS   + (size_t)(c + 1) * KC) * 2, KC * 2);
                __builtin_amdgcn_s_wait_tensorcnt(2);  // chunk c done; c+1 may be in flight
            } else {
                __builtin_amdgcn_s_wait_tensorcnt(0);  // last chunk done
            }
        }
        __syncthreads();                                // data visible to all 8 waves

        const float* P = sPos[cur];
        const float* L = sLog[cur];
#pragma unroll 4
        for (int i = 0; i < KC; ++i) {
            const float dx = yx - P[3 * i + 0];
            const float dy = yy - P[3 * i + 1];
            const float dz = yz - P[3 * i + 2];
            const float d  = dx * dx + dy * dy + dz * dz;
            if (d < bd2) {                 // insert into running top-3 (smallest)
                const float l0 = L[2 * i + 0], l1 = L[2 * i + 1];
                if (d < bd0) {
                    bd2 = bd1; g20 = g10; g21 = g11;
                    bd1 = bd0; g10 = g00; g11 = g01;
                    bd0 = d;   g00 = l0;  g01 = l1;
                } else if (d < bd1) {
                    bd2 = bd1; g20 = g10; g21 = g11;
                    bd1 = d;   g10 = l0;  g11 = l1;
                } else {
                    bd2 = d;   g20 = l0;  g21 = l1;
                }
            }
        }
        __syncthreads();                                // buffer may be overwritten next iter
    }

    const float w0 = 1.f / fmaxf(bd0, 1e-16f);
    const float w1 = 1.f / fmaxf(bd1, 1e-16f);
    const float w2 = 1.f / fmaxf(bd2, 1e-16f);
    const float inv = 1.f / (w0 + w1 + w2);
    out[(size_t)(b * NPTS + q) * 2 + 0] = (w0 * g00 + w1 * g10 + w2 * g20) * inv;
    out[(size_t)(b * NPTS + q) * 2 + 1] = (w0 * g01 + w1 * g11 + w2 * g21) * inv;
}

// ---------------------------------------------------------------------
extern "C" void kernel_launch(void* const* d_in, const int* in_sizes, int n_in,
                              void* d_out, int out_size, void* d_ws, size_t ws_size,
                              hipStream_t stream)
{
    // Resolve inputs by unique flat sizes (robust to pytree flattening order).
    auto findsz = [&](int sz) -> int {
        for (int i = 0; i < n_in; ++i) if (in_sizes[i] == sz) return i;
        return 0;
    };
    const float* x   = (const float*)d_in[findsz(BB * NPTS * CFEAT)]; // 1179648
    const float* pos = (const float*)d_in[findsz(BB * NPTS * 3)];     // 393216
    const int ibl = findsz(2);                    // lin bias -> lin = (Wl, bl)
    const float* Wl = (const float*)d_in[ibl - 1];
    const float* bl = (const float*)d_in[ibl];
    const int i1 = findsz(12 * 64);               // mlp1: W0,b0,W1,b1
    const float* W10 = (const float*)d_in[i1],     *b10 = (const float*)d_in[i1 + 1];
    const float* W11 = (const float*)d_in[i1 + 2], *b11 = (const float*)d_in[i1 + 3];
    const int i2 = findsz(128 * 256);             // mlp2 W1 (unique 32768)
    const float* W20 = (const float*)d_in[i2 - 2], *b20 = (const float*)d_in[i2 - 1];
    const float* W21 = (const float*)d_in[i2],     *b21 = (const float*)d_in[i2 + 1];
    const int i3 = findsz(320 * 128);             // mlp3 W0 (unique 40960)
    const float* W30 = (const float*)d_in[i3],     *b30 = (const float*)d_in[i3 + 1];
    const float* W31 = (const float*)d_in[i3 + 2], *b31 = (const float*)d_in[i3 + 3];

    // Workspace layout: f1 (8 MB) | ctx (8 KB) | logits (256 KB)
    char* ws = (char*)d_ws;
    float*    f1_ws  = (float*)ws;
    unsigned* ctx    = (unsigned*)(ws + (size_t)BB * SS * 64 * sizeof(float));
    float*    logits = (float*)(ws + (size_t)BB * SS * 64 * sizeof(float)
                                   + (size_t)BB * 256 * sizeof(unsigned));
    float* out = (float*)d_out;

    const int tiles  = (BB * SS) / 16;   // 2048 16-point tiles
    const int blocks = tiles / 2;        // 2 waves per block

    init_ctx_kernel<<<(BB * 256 + 255) / 256, 256, 0, stream>>>(ctx, BB * 256);
    encoder_kernel<<<blocks, 64, 0, stream>>>(x, pos, W10, b10, W11, b11,
                                              W20, b20, W21, b21, f1_ws, ctx);
    head_kernel<<<blocks, 64, 0, stream>>>(f1_ws, ctx, W30, b30, W31, b31,
                                           Wl, bl, logits);
    knn_kernel<<<BB * (NPTS / 256), 256, 0, stream>>>(pos, logits, out);
}